// ArrowLoraLinearLayer_49503793054546
// MI455X (gfx1250) — compile-verified
//
#include <hip/hip_runtime.h>

typedef __attribute__((ext_vector_type(2))) float v2f;
typedef __attribute__((ext_vector_type(8))) float v8f;
typedef int v4i __attribute__((vector_size(16)));

#define F_IN  2048
#define N_EXP 8
#define OUT_F 2048

#define BM 128
#define BN 64
#define BK 32
#define LDSS 36   // padded LDS row stride (floats): 16B-aligned rows, conflict-free b64 reads
#define NKB (F_IN / BK)

// ---- async global -> LDS copy (CDNA5 ASYNCcnt path), with sync fallback ----
#if __has_builtin(__builtin_amdgcn_global_load_async_to_lds_b128)
#define HAVE_ASYNC_LDS 1
#else
#define HAVE_ASYNC_LDS 0
#endif

typedef __attribute__((address_space(1))) v4i* as1_v4i_ptr;
typedef __attribute__((address_space(3))) v4i* as3_v4i_ptr;

__device__ __forceinline__ void cp16_to_lds(const float* g, float* l)
{
#if HAVE_ASYNC_LDS
    __builtin_amdgcn_global_load_async_to_lds_b128(
        (as1_v4i_ptr)(void*)g, (as3_v4i_ptr)(void*)l, 0, 0);
#else
    *(float4*)l = *(const float4*)g;
#endif
}

__device__ __forceinline__ void async_wait0()
{
#if HAVE_ASYNC_LDS
#if __has_builtin(__builtin_amdgcn_s_wait_asynccnt)
    __builtin_amdgcn_s_wait_asynccnt(0);
#else
    asm volatile("s_wait_asynccnt 0" ::: "memory");
#endif
#endif
}

// ---------------------------------------------------------------------------
// Routing: sim = |tok . proto_e|, top-2, softmax over the kept pair.
// One wave (32 lanes) per token; prototypes staged in LDS.
// ---------------------------------------------------------------------------
__global__ __launch_bounds__(256) void arrow_route(const float* __restrict__ x,
                                                   const float* __restrict__ protos,
                                                   float* __restrict__ coeff)
{
    __shared__ float ps[N_EXP * F_IN];                 // 64 KB
    const int tid = threadIdx.x;

    const float4* pv = (const float4*)protos;
    float4* sv = (float4*)ps;
    for (int i = tid; i < N_EXP * F_IN / 4; i += 256) sv[i] = pv[i];
    __syncthreads();

    const int lane  = tid & 31;
    const int wave  = tid >> 5;
    const int token = blockIdx.x * 8 + wave;

    float acc[N_EXP];
#pragma unroll
    for (int e = 0; e < N_EXP; ++e) acc[e] = 0.0f;

    const float* xrow = x + (size_t)token * F_IN;
    for (int i = lane; i < F_IN; i += 32) {
        float xv = xrow[i];
#pragma unroll
        for (int e = 0; e < N_EXP; ++e)
            acc[e] = fmaf(xv, ps[e * F_IN + i], acc[e]);
    }

    // wave32 butterfly reduction
#pragma unroll
    for (int off = 16; off > 0; off >>= 1) {
#pragma unroll
        for (int e = 0; e < N_EXP; ++e)
            acc[e] += __shfl_xor(acc[e], off, 32);
    }

#pragma unroll
    for (int e = 0; e < N_EXP; ++e) acc[e] = fabsf(acc[e]);

    // top-2
    int i1 = 0; float v1 = acc[0];
#pragma unroll
    for (int e = 1; e < N_EXP; ++e) if (acc[e] > v1) { v1 = acc[e]; i1 = e; }
    int i2 = -1; float v2 = -__builtin_inff();
#pragma unroll
    for (int e = 0; e < N_EXP; ++e) if (e != i1 && acc[e] > v2) { v2 = acc[e]; i2 = e; }

    // softmax over {v1, v2} (temperature = 1); everything else exactly 0
    float w2  = __expf(v2 - v1);
    float inv = 1.0f / (1.0f + w2);

    if (lane < N_EXP) {
        float c = (lane == i1) ? inv : ((lane == i2) ? (w2 * inv) : 0.0f);
        coeff[(size_t)token * N_EXP + lane] = c;
    }
}

// ---------------------------------------------------------------------------
// Main GEMM: delta[t,o] = scaling * sum_e coeff[t,e] * sum_i tok[t,i]*W[e,o,i]
// Block tile 128(M) x 64(N), 8 wave32 waves; fp32 WMMA 16x16x4.
// A tile staged once per K-block; B (W-slice) tile double-buffered with
// async global->LDS copies issued for tile i+1 while tile i computes.
// ---------------------------------------------------------------------------
__global__ __launch_bounds__(256) void arrow_gemm(const float* __restrict__ x,
                                                  const float* __restrict__ W,
                                                  const float* __restrict__ coeff,
                                                  const int* __restrict__ scaling,
                                                  float* __restrict__ out)
{
    __shared__ float As[BM * LDSS];        // token tile (128 x 32, padded)
    __shared__ float Bs[2][BN * LDSS];     // W tile ping-pong (64 x 32, padded)

    const int tid   = threadIdx.x;
    const int lane  = tid & 31;
    const int wave  = tid >> 5;
    const int Nbase = blockIdx.x * BN;
    const int Mbase = blockIdx.y * BM;

    const int ml = lane & 15;   // row-within-16 (A/B fragment M/N index)
    const int kh = lane >> 4;   // selects K pair {0,1} vs {2,3}

    // per-lane routing coefficients for this wave's 16-token strip
    const int mrow = Mbase + wave * 16 + ml;
    float cf[N_EXP];
#pragma unroll
    for (int e = 0; e < N_EXP; ++e) cf[e] = coeff[(size_t)mrow * N_EXP + e];

    v8f acc[4] = {};

    // ---- staging helpers (per-lane 16B chunks) ----
    // A tile: 128 rows x 8 float4  -> 4 chunks/thread
    // B tile:  64 rows x 8 float4  -> 2 chunks/thread
#define STAGE_A(KB)                                                          \
    {                                                                        \
        _Pragma("unroll")                                                    \
        for (int it = 0; it < (BM * 8) / 256; ++it) {                        \
            int f = tid + it * 256;                                          \
            int row = f >> 3, c4 = f & 7;                                    \
            cp16_to_lds(&x[(size_t)(Mbase + row) * F_IN + (KB) * BK + c4*4], \
                        &As[row * LDSS + c4 * 4]);                           \
        }                                                                    \
    }
#define STAGE_B(E, KB, BUF)                                                  \
    {                                                                        \
        _Pragma("unroll")                                                    \
        for (int it = 0; it < (BN * 8) / 256; ++it) {                        \
            int f = tid + it * 256;                                          \
            int row = f >> 3, c4 = f & 7;                                    \
            cp16_to_lds(&W[(size_t)(E) * OUT_F * F_IN +                      \
                           (size_t)(Nbase + row) * F_IN + (KB) * BK + c4*4], \
                        &Bs[BUF][row * LDSS + c4 * 4]);                      \
        }                                                                    \
    }

    // prologue: first B tile in flight
    STAGE_B(0, 0, 0);
    int bufB = 0;

    for (int kb = 0; kb < NKB; ++kb) {
        // stage A for this K-block (previous iteration's barrier freed As)
        STAGE_A(kb);
        async_wait0();
        __syncthreads();     // As + Bs[bufB] for (kb, e=0) now visible

        for (int e = 0; e < N_EXP; ++e) {
            // issue async copy of the NEXT tile into the other B buffer
            int en = e + 1, kbn = kb;
            if (en == N_EXP) { en = 0; ++kbn; }
            const bool have_next = (kbn < NKB);
            if (have_next) STAGE_B(en, kbn, bufB ^ 1);

            // ---- compute on As / Bs[bufB] ----
            const float c = cf[e];
#pragma unroll
            for (int k0 = 0; k0 < BK; k0 += 4) {
                // A frag: lane l<16 -> A[M=l][k0..k0+1], l>=16 -> A[M=l-16][k0+2..k0+3]
                v2f a = *(const v2f*)&As[(wave * 16 + ml) * LDSS + k0 + 2 * kh];
                a.x *= c; a.y *= c;
#pragma unroll
                for (int n = 0; n < 4; ++n) {
                    v2f b = *(const v2f*)&Bs[bufB][(n * 16 + ml) * LDSS + k0 + 2 * kh];
                    acc[n] = __builtin_amdgcn_wmma_f32_16x16x4_f32(
                        false, a, false, b, (short)0, acc[n], false, false);
                }
            }

            if (have_next) async_wait0();   // our async writes to Bs[bufB^1] landed
            __syncthreads();                // publish next tile / release current
            if (have_next) bufB ^= 1;
        }
    }

    // epilogue: D layout -> VGPR r: lanes 0-15 (M=r, N=lane), lanes 16-31 (M=r+8, N=lane-16)
    const float scale = (float)(*scaling);
    const int trow = Mbase + wave * 16 + 8 * kh;
    const int ocol = Nbase + ml;
#pragma unroll
    for (int n = 0; n < 4; ++n) {
#pragma unroll
        for (int r = 0; r < 8; ++r) {
            out[(size_t)(trow + r) * OUT_F + (ocol + n * 16)] = scale * acc[n][r];
        }
    }
#undef STAGE_A
#undef STAGE_B
}

// ---------------------------------------------------------------------------
extern "C" void kernel_launch(void* const* d_in, const int* in_sizes, int n_in,
                              void* d_out, int out_size, void* d_ws, size_t ws_size,
                              hipStream_t stream)
{
    const float* x       = (const float*)d_in[0];
    const float* protos  = (const float*)d_in[1];
    const float* W       = (const float*)d_in[2];
    const int*   scaling = (const int*)d_in[3];
    float*       out     = (float*)d_out;
    float*       coeff   = (float*)d_ws;   // T x 8 floats

    const int T = in_sizes[0] / F_IN;      // 8192 tokens

    arrow_route<<<dim3(T / 8), dim3(256), 0, stream>>>(x, protos, coeff);
    arrow_gemm<<<dim3(OUT_F / BN, T / BM), dim3(256), 0, stream>>>(x, W, coeff, scaling, out);
}